// MLFFNet_36773509989087
// MI455X (gfx1250) — compile-verified
//
#include <hip/hip_runtime.h>
#include <hip/hip_bf16.h>

// ---------------------------------------------------------------------------
// MLFF net for MI455X (gfx1250, wave32).
//   K1: per-atom MLP fwd + analytic dE/dx via V_WMMA_F32_16X16X4_F32
//       (one wave per 16-atom tile; LDS handoff between WMMA stages).
//       K-loop bodies are branch-free: full chunks use unconditional loads,
//       tails use clamped-address loads + v_cndmask value masking.
//   K2: force contraction -- bandwidth-bound stream of dfeat (413 MB),
//       dE rows gathered once into LDS per (b,n), barrier-free hot loop.
//   K3: Etot reduction.
// ---------------------------------------------------------------------------

typedef __attribute__((ext_vector_type(2))) float v2f;
typedef __attribute__((ext_vector_type(8))) float v8f;

#define BB 4
#define NN 2048
#define MM 100
#define FF 42
#define H1 64
#define H2 32

__device__ __forceinline__ v8f wmma_k4(v2f a, v2f b, v8f c) {
  // 8 args: (neg_a, A, neg_b, B, c_mod, C, reuse_a, reuse_b)
  return __builtin_amdgcn_wmma_f32_16x16x4_f32(
      false, a, false, b, (short)0, c, false, false);
}

__device__ __forceinline__ float softplus_f(float x) {
  return fmaxf(x, 0.0f) + log1pf(__expf(-fabsf(x)));
}
__device__ __forceinline__ float sigmoid_f(float x) {
  return 1.0f / (1.0f + __expf(-x));
}

// ---------------------------------------------------------------------------
// Kernel 1: MLP forward + gradient chain.  One wave (32 threads) per 16 atoms.
// Writes Ei [B*N] (into d_out) and dE [B*N, F] (into workspace).
// ---------------------------------------------------------------------------
__global__ __launch_bounds__(32)
void mlp_kernel(const float* __restrict__ image,
                const float* __restrict__ W0, const float* __restrict__ b0,
                const float* __restrict__ W1, const float* __restrict__ b1,
                const float* __restrict__ W2, const float* __restrict__ b2,
                float* __restrict__ Ei_out,
                float* __restrict__ dE)
{
  __shared__ float sL0[16][H1 + 4];   // softplus(z0)   (stride 68: conflict-free)
  __shared__ float sD0[16][H1 + 4];   // sigmoid(z0)
  __shared__ float sL1[16][H2 + 4];   // softplus(z1)
  __shared__ float sGA[16][H2 + 4];   // d1 * W2
  __shared__ float sGH[16][H1 + 4];   // d0 * ((d1*W2) @ W1)

  const int lane = threadIdx.x;       // 0..31
  const int lidx = lane & 15;         // A: row M;  B: col N;  C/D: col N
  const int half = lane >> 4;         // selects K pair (A/B) / row offset (C/D)
  const int atom0 = blockIdx.x * 16;  // flat over B*N

  const float* imgRow = image + (size_t)(atom0 + lidx) * FF;

  // ---- z0 = image @ W0^T ------------------------------------------- WMMA
  v8f acc0[4] = {};
  // full K chunks: kb = 0..36 -> k0+1 <= 39 < 42, unconditional loads
  for (int kb = 0; kb < 40; kb += 4) {
    const int k0 = kb + 2 * half;
    v2f a; a.x = imgRow[k0]; a.y = imgRow[k0 + 1];
#pragma unroll
    for (int nb = 0; nb < 4; ++nb) {
      const int n = nb * 16 + lidx;
      v2f b; b.x = W0[n * FF + k0]; b.y = W0[n * FF + k0 + 1];
      acc0[nb] = wmma_k4(a, b, acc0[nb]);
    }
  }
  {
    // tail chunk kb=40: half==1 lanes (k=42,43) are zero.  Load from a
    // clamped always-valid address, then v_cndmask the value (no exec branch).
    const int ktc = half ? 0 : 40;
    const float m = half ? 0.0f : 1.0f;
    v2f a;
    {
      const float t0 = imgRow[ktc], t1 = imgRow[ktc + 1];
      a.x = t0 * m; a.y = t1 * m;
    }
#pragma unroll
    for (int nb = 0; nb < 4; ++nb) {
      const int n = nb * 16 + lidx;
      const float u0 = W0[n * FF + ktc], u1 = W0[n * FF + ktc + 1];
      v2f b; b.x = u0 * m; b.y = u1 * m;
      acc0[nb] = wmma_k4(a, b, acc0[nb]);
    }
  }
  // activations, stage L0/d0 to LDS
#pragma unroll
  for (int nb = 0; nb < 4; ++nb) {
    const int h = nb * 16 + lidx;
    const float bias = b0[h];
#pragma unroll
    for (int r = 0; r < 8; ++r) {
      const int row = r + 8 * half;
      const float z = acc0[nb][r] + bias;
      sL0[row][h] = softplus_f(z);
      sD0[row][h] = sigmoid_f(z);
    }
  }
  __syncthreads();

  // ---- z1 = L0 @ W1^T  (K=64, no tail) ----------------------------- WMMA
  v8f acc1[2] = {};
  for (int kb = 0; kb < H1; kb += 4) {
    const int k0 = kb + 2 * half;
    v2f a; a.x = sL0[lidx][k0]; a.y = sL0[lidx][k0 + 1];
#pragma unroll
    for (int nb = 0; nb < 2; ++nb) {
      const int n = nb * 16 + lidx;
      v2f b; b.x = W1[n * H1 + k0]; b.y = W1[n * H1 + k0 + 1];
      acc1[nb] = wmma_k4(a, b, acc1[nb]);
    }
  }
  // L1 (for Ei) and gA = sigmoid(z1) * W2 (grad seed)
#pragma unroll
  for (int nb = 0; nb < 2; ++nb) {
    const int h = nb * 16 + lidx;
    const float bias = b1[h];
    const float w2h  = W2[h];
#pragma unroll
    for (int r = 0; r < 8; ++r) {
      const int row = r + 8 * half;
      const float z = acc1[nb][r] + bias;
      sL1[row][h] = softplus_f(z);
      sGA[row][h] = sigmoid_f(z) * w2h;
    }
  }
  __syncthreads();

  // ---- Ei = L1 @ W2^T + b2 (skinny: plain VALU, lanes 0..15) ------------
  if (half == 0) {
    float s = b2[0];
#pragma unroll
    for (int h = 0; h < H2; ++h) s += sL1[lidx][h] * W2[h];
    Ei_out[atom0 + lidx] = s;
  }

  // ---- gg = gA @ W1  (B[k][n] = W1[k*H1+n], K=32, no tail) --------- WMMA
  v8f accg[4] = {};
  for (int kb = 0; kb < H2; kb += 4) {
    const int k0 = kb + 2 * half;
    v2f a; a.x = sGA[lidx][k0]; a.y = sGA[lidx][k0 + 1];
#pragma unroll
    for (int nb = 0; nb < 4; ++nb) {
      const int n = nb * 16 + lidx;
      v2f b; b.x = W1[k0 * H1 + n]; b.y = W1[(k0 + 1) * H1 + n];
      accg[nb] = wmma_k4(a, b, accg[nb]);
    }
  }
  // gH = d0 * gg, stage
#pragma unroll
  for (int nb = 0; nb < 4; ++nb) {
    const int h = nb * 16 + lidx;
#pragma unroll
    for (int r = 0; r < 8; ++r) {
      const int row = r + 8 * half;
      sGH[row][h] = sD0[row][h] * accg[nb][r];
    }
  }
  __syncthreads();

  // ---- dE = gH @ W0  (B[k][n] = W0[k*FF+n], N tail at nb=2) -------- WMMA
  v8f accd[3] = {};
  for (int kb = 0; kb < H1; kb += 4) {
    const int k0 = kb + 2 * half;
    v2f a; a.x = sGH[lidx][k0]; a.y = sGH[lidx][k0 + 1];
    // nb = 0,1: n = 0..31 < 42, unconditional
#pragma unroll
    for (int nb = 0; nb < 2; ++nb) {
      const int n = nb * 16 + lidx;
      v2f b; b.x = W0[k0 * FF + n]; b.y = W0[(k0 + 1) * FF + n];
      accd[nb] = wmma_k4(a, b, accd[nb]);
    }
    // nb = 2: n = 32..47, mask n >= 42 via clamped address + value select
    {
      const int n  = 32 + lidx;
      const int nc = (n < FF) ? n : 0;        // v_cndmask on index
      const float m = (n < FF) ? 1.0f : 0.0f;
      const float u0 = W0[k0 * FF + nc], u1 = W0[(k0 + 1) * FF + nc];
      v2f b; b.x = u0 * m; b.y = u1 * m;
      accd[2] = wmma_k4(a, b, accd[2]);
    }
  }
#pragma unroll
  for (int nb = 0; nb < 3; ++nb) {
    const int f = nb * 16 + lidx;
    if (f < FF) {
#pragma unroll
      for (int r = 0; r < 8; ++r) {
        const int row = r + 8 * half;
        dE[(size_t)(atom0 + row) * FF + f] = accd[nb][r];
      }
    }
  }
}

// ---------------------------------------------------------------------------
// Kernel 2: force[b,n,:] = sum_m sum_f dE[b, nbr(b,n,m), f] * dfeat[b,n,m,f,:]
// One 128-thread block per (b,n).  Gathered dE rows staged once in LDS; then
// a barrier-free coalesced stream of 50.4 KB of dfeat per block.
// ---------------------------------------------------------------------------
__global__ __launch_bounds__(128)
void force_kernel(const float* __restrict__ dfeat,
                  const int* __restrict__ neighbor,
                  const float* __restrict__ dE,
                  float* __restrict__ force)
{
  __shared__ float sdE[MM * FF];   // 4200 floats = 16.8 KB
  __shared__ int   sIdx[MM];
  __shared__ float sRed[FF * 3];

  const int p  = threadIdx.x;
  const int bn = blockIdx.x;       // flat b*N + n
  const int b  = bn / NN;

  if (p < MM) sIdx[p] = neighbor[(size_t)bn * MM + p];   // 1-based; 0 = none
  __syncthreads();

  for (int t = p; t < MM * FF; t += 128) {
    const int mm = t / FF;
    const int f  = t - mm * FF;
    const int nb = sIdx[mm];
    const int ic = (nb > 0) ? (nb - 1) : 0;              // clamped gather index
    const float v = dE[((size_t)b * NN + ic) * FF + f];  // unconditional load
    sdE[t] = (nb > 0) ? v : 0.0f;                        // value mask
  }
  __syncthreads();

  if (p < FF * 3) {
    const float* drow = dfeat + (size_t)bn * MM * (FF * 3) + p;
    const int f = p / 3;
    float acc = 0.0f;
#pragma unroll 4
    for (int m = 0; m < MM; ++m)
      acc += drow[(size_t)m * (FF * 3)] * sdE[m * FF + f];
    sRed[p] = acc;
  }
  __syncthreads();

  if (p < 3) {
    float s = 0.0f;
    for (int j = 0; j < FF; ++j) s += sRed[p + 3 * j];
    force[(size_t)bn * 3 + p] = s;
  }
}

// ---------------------------------------------------------------------------
// Kernel 3: Etot[b] = sum_n Ei[b,n]
// ---------------------------------------------------------------------------
__global__ __launch_bounds__(256)
void etot_kernel(const float* __restrict__ Ei, float* __restrict__ Etot)
{
  __shared__ float sred[256];
  const int b = blockIdx.x;
  float s = 0.0f;
  for (int i = threadIdx.x; i < NN; i += 256) s += Ei[(size_t)b * NN + i];
  sred[threadIdx.x] = s;
  __syncthreads();
  for (int st = 128; st > 0; st >>= 1) {
    if ((int)threadIdx.x < st) sred[threadIdx.x] += sred[threadIdx.x + st];
    __syncthreads();
  }
  if (threadIdx.x == 0) Etot[b] = sred[0];
}

// ---------------------------------------------------------------------------
extern "C" void kernel_launch(void* const* d_in, const int* in_sizes, int n_in,
                              void* d_out, int out_size, void* d_ws, size_t ws_size,
                              hipStream_t stream) {
  const float* image    = (const float*)d_in[0];
  const float* dfeat    = (const float*)d_in[1];
  // d_in[2] Egroup_weight, d_in[3] divider: unused by reference outputs
  const float* W0       = (const float*)d_in[4];
  const float* b0       = (const float*)d_in[5];
  const float* W1       = (const float*)d_in[6];
  const float* b1       = (const float*)d_in[7];
  const float* W2       = (const float*)d_in[8];
  const float* b2       = (const float*)d_in[9];
  const int*   neighbor = (const int*)d_in[10];

  float* out   = (float*)d_out;
  float* Etot  = out;                 // [B]
  float* Ei    = out + BB;            // [B*N]
  float* force = out + BB + BB * NN;  // [B*N*3]
  float* dEws  = (float*)d_ws;        // [B*N*F] scratch (1.4 MB)

  mlp_kernel<<<(BB * NN) / 16, 32, 0, stream>>>(image, W0, b0, W1, b1, W2, b2,
                                                Ei, dEws);
  force_kernel<<<BB * NN, 128, 0, stream>>>(dfeat, neighbor, dEws, force);
  etot_kernel<<<BB, 256, 0, stream>>>(Ei, Etot);
}